// DeformableDETR_Demo_11484742549697
// MI455X (gfx1250) — compile-verified
//
#include <hip/hip_runtime.h>
#include <hip/hip_bf16.h>
#include <math.h>

typedef float v2f __attribute__((ext_vector_type(2)));
typedef float v8f __attribute__((ext_vector_type(8)));

// ---------------------------------------------------------------------------
// LDS-tiled transpose:  D[N,K] = S[K,N]^T   (fp32)
// ---------------------------------------------------------------------------
__global__ __launch_bounds__(256) void transpose_f32(
    const float* __restrict__ S, float* __restrict__ D, int R, int Ccols)
{
    __shared__ float tile[32][33];
    const int c0 = blockIdx.x * 32, r0 = blockIdx.y * 32;
    const int tx = threadIdx.x & 31, ty = threadIdx.x >> 5;   // 32 x 8
#pragma unroll
    for (int i = 0; i < 32; i += 8) {
        const int r = r0 + ty + i, c = c0 + tx;
        tile[ty + i][tx] = (r < R && c < Ccols) ? S[(size_t)r * Ccols + c] : 0.0f;
    }
    __syncthreads();
#pragma unroll
    for (int i = 0; i < 32; i += 8) {
        const int r = c0 + ty + i, c = r0 + tx;               // D is [Ccols, R]
        if (r < Ccols && c < R) D[(size_t)r * R + c] = tile[tx][ty + i];
    }
}

// ---------------------------------------------------------------------------
// C[M,N] = A[M,256] @ B[256,N] + bias[N], with B supplied TRANSPOSED (BT[N,256]).
// One wave computes a 32x16 tile (two 16x16 WMMA accumulators sharing the B
// fragment). K=256 and N are compile-time constants so every inner-loop access
// is base-pointer + immediate offset.
// Fragment layouts per CDNA5 ISA 7.12.2 (V_WMMA_F32_16X16X4_F32):
//   A (16x4): lane<16 -> row=lane,    {K=k0,k0+1}; lane>=16 -> row=lane-16, {k0+2,k0+3}
//   B (4x16): v0=B[k0+2*hi][n], v1=B[k0+2*hi+1][n]  ==  BT[n][k0+2*hi..+1] (contiguous)
//   C/D: element r of lane -> row r + 8*hi, col lane&15
// ---------------------------------------------------------------------------
template <int N>
__global__ __launch_bounds__(128) void gemm_f32_wmma(
    const float* __restrict__ A, const float* __restrict__ BT,
    const float* __restrict__ bias, float* __restrict__ C, int M)
{
    constexpr int K = 256;
    const int lane = threadIdx.x & 31;
    const int wave = threadIdx.x >> 5;
    const int nt   = blockIdx.y * 4 + wave;       // wave-uniform
    if (nt * 16 >= N) return;

    const int n0 = nt * 16;
    const int hi = lane >> 4;                     // 0: lanes 0-15, 1: lanes 16-31
    const int nl = lane & 15;
    const int m0 = blockIdx.x * 32;               // two stacked M-tiles per wave

    int ar0 = m0 + nl;       if (ar0 > M - 1) ar0 = M - 1;   // clamp loads
    int ar1 = m0 + 16 + nl;  if (ar1 > M - 1) ar1 = M - 1;

    const float* __restrict__ Ap0 = A  + (size_t)ar0 * K + 2 * hi;
    const float* __restrict__ Ap1 = A  + (size_t)ar1 * K + 2 * hi;
    const float* __restrict__ Bp  = BT + (size_t)(n0 + nl) * K + 2 * hi;

    v8f acc0 = {}, acc1 = {};
#pragma unroll 8
    for (int k0 = 0; k0 < K; k0 += 4) {
        const v2f b  = *(const v2f*)(Bp  + k0);   // contiguous b64, immediate offset
        const v2f a0 = *(const v2f*)(Ap0 + k0);
        const v2f a1 = *(const v2f*)(Ap1 + k0);
        acc0 = __builtin_amdgcn_wmma_f32_16x16x4_f32(
            false, a0, false, b, (short)0, acc0, false, false);
        acc1 = __builtin_amdgcn_wmma_f32_16x16x4_f32(
            false, a1, false, b, (short)0, acc1, false, false);
    }

    const float bv = bias[n0 + nl];
#pragma unroll
    for (int r = 0; r < 8; ++r) {
        const int row0 = m0 + 8 * hi + r;
        const int row1 = row0 + 16;
        if (row0 < M) C[(size_t)row0 * N + n0 + nl] = acc0[r] + bv;
        if (row1 < M) C[(size_t)row1 * N + n0 + nl] = acc1[r] + bv;
    }
}

// ---------------------------------------------------------------------------
// Multi-scale deformable sampling.
// Block = 256 threads = 8 waves; block -> query, wave -> head, lane -> channel.
// Lanes 0..15 own the 16 (level,point) slots: softmax weight + pixel coords,
// computed once with __shfl_xor reductions, then broadcast in the point loop.
// Corner loads: 32 lanes * 4B consecutive = one coalesced 128B transaction.
// ---------------------------------------------------------------------------
__global__ __launch_bounds__(256) void ms_deform_sample(
    const float* __restrict__ refp,     // [L,2]
    const float* __restrict__ feat,     // [L,256] input_flatten
    const int*   __restrict__ shapes,   // [4][2] = (H,W)
    const int*   __restrict__ lstart,   // [4]
    const float* __restrict__ off,      // [L,256]
    const float* __restrict__ attnlog,  // [L,128]
    float*       __restrict__ mid,      // [L,256]
    int L)
{
    const int q    = blockIdx.x;
    const int h    = threadIdx.x >> 5;  // head
    const int lane = threadIdx.x & 31;  // channel within head
    if (q >= L) return;

    const float rx = refp[2 * q + 0];
    const float ry = refp[2 * q + 1];

    // --- softmax over the 16 (level,point) logits of this head -------------
    float logit = -3.0e38f;
    if (lane < 16) logit = attnlog[(size_t)q * 128 + h * 16 + lane];
    float mx = logit;
#pragma unroll
    for (int i = 1; i < 16; i <<= 1) mx = fmaxf(mx, __shfl_xor(mx, i, 32));
    float e = (lane < 16) ? __expf(logit - mx) : 0.0f;
    float s = e;
#pragma unroll
    for (int i = 1; i < 16; i <<= 1) s += __shfl_xor(s, i, 32);

    // --- per-slot pixel coordinates (lanes 0..15) --------------------------
    float wgt = 0.0f, x = 0.0f, y = 0.0f;
    if (lane < 16) {
        wgt = e / s;
        const int lvl = lane >> 2;
        const int pnt = lane & 3;
        const int H = shapes[2 * lvl + 0];
        const int W = shapes[2 * lvl + 1];
        const size_t ob = (size_t)q * 256 + (size_t)(((h * 4 + lvl) * 4 + pnt) * 2);
        const float ox = off[ob + 0];
        const float oy = off[ob + 1];
        const float lx = rx + ox / (float)W;   // match reference rounding
        const float ly = ry + oy / (float)H;
        x = lx * (float)W - 0.5f;
        y = ly * (float)H - 0.5f;
    }

    // --- accumulate 16 bilinear samples ------------------------------------
    const float* __restrict__ fb = feat + h * 32 + lane;
    float acc = 0.0f;
#pragma unroll
    for (int p = 0; p < 16; ++p) {
        const float xp = __shfl(x, p, 32);
        const float yp = __shfl(y, p, 32);
        const float wp = __shfl(wgt, p, 32);
        const int lvl = p >> 2;
        const int H  = shapes[2 * lvl + 0];
        const int W  = shapes[2 * lvl + 1];
        const int st = lstart[lvl];

        const float x0f = floorf(xp), y0f = floorf(yp);
        const float wx1 = xp - x0f,   wy1 = yp - y0f;
        const float wx0 = 1.0f - wx1, wy0 = 1.0f - wy1;
        const int x0 = (int)x0f, y0 = (int)y0f;
        const int x1 = x0 + 1,   y1 = y0 + 1;

        const bool vx0 = (x0 >= 0) & (x0 < W);
        const bool vx1 = (x1 >= 0) & (x1 < W);
        const bool vy0 = (y0 >= 0) & (y0 < H);
        const bool vy1 = (y1 >= 0) & (y1 < H);

        float v00 = 0.f, v10 = 0.f, v01 = 0.f, v11 = 0.f;
        if (vx0 & vy0) v00 = fb[(size_t)(st + y0 * W + x0) * 256];
        if (vx1 & vy0) v10 = fb[(size_t)(st + y0 * W + x1) * 256];
        if (vx0 & vy1) v01 = fb[(size_t)(st + y1 * W + x0) * 256];
        if (vx1 & vy1) v11 = fb[(size_t)(st + y1 * W + x1) * 256];

        acc += wp * (v00 * (wx0 * wy0) + v10 * (wx1 * wy0) +
                     v01 * (wx0 * wy1) + v11 * (wx1 * wy1));
    }

    mid[(size_t)q * 256 + h * 32 + lane] = acc;
}

// ---------------------------------------------------------------------------
extern "C" void kernel_launch(void* const* d_in, const int* in_sizes, int n_in,
                              void* d_out, int out_size, void* d_ws, size_t ws_size,
                              hipStream_t stream)
{
    const float* query  = (const float*)d_in[0];
    const float* refp   = (const float*)d_in[1];
    const float* feat   = (const float*)d_in[2];
    const int*   shapes = (const int*)  d_in[3];
    const int*   lstart = (const int*)  d_in[4];
    const float* W_off  = (const float*)d_in[5];
    const float* b_off  = (const float*)d_in[6];
    const float* W_attn = (const float*)d_in[7];
    const float* b_attn = (const float*)d_in[8];
    const float* W_out  = (const float*)d_in[9];
    const float* b_out  = (const float*)d_in[10];
    float* out = (float*)d_out;

    const int L = in_sizes[1] / 2;   // reference_points is [1, L, 2]

    // Workspace: off[L,256] | attn[L,128] | mid[L,256] | WT_off[256,256]
    //            | WT_attn[128,256] | WT_out[256,256]     (~32 MB total)
    float* off    = (float*)d_ws;
    float* attn   = off    + (size_t)L * 256;
    float* mid    = attn   + (size_t)L * 128;
    float* WToff  = mid    + (size_t)L * 256;
    float* WTattn = WToff  + (size_t)256 * 256;
    float* WTout  = WTattn + (size_t)128 * 256;

    // --- weight transposes (W[K=256,N] -> WT[N,256]) -----------------------
    transpose_f32<<<dim3(256 / 32, 256 / 32), dim3(256), 0, stream>>>(W_off,  WToff,  256, 256);
    transpose_f32<<<dim3(128 / 32, 256 / 32), dim3(256), 0, stream>>>(W_attn, WTattn, 256, 128);
    transpose_f32<<<dim3(256 / 32, 256 / 32), dim3(256), 0, stream>>>(W_out,  WTout,  256, 256);

    const int MtPairs = (L + 31) / 32;           // 32 rows per wave-tile
    const dim3 blk(128);

    // offsets = query @ W_off + b_off            [L,256]
    gemm_f32_wmma<256><<<dim3(MtPairs, 4), blk, 0, stream>>>(query, WToff, b_off, off, L);
    // attn logits = query @ W_attn + b_attn      [L,128]
    gemm_f32_wmma<128><<<dim3(MtPairs, 2), blk, 0, stream>>>(query, WTattn, b_attn, attn, L);
    // softmax + multi-scale bilinear sampling    [L,256]
    ms_deform_sample<<<dim3(L), dim3(256), 0, stream>>>(refp, feat, shapes, lstart, off, attn, mid, L);
    // out = mid @ W_out + b_out                  [L,256]
    gemm_f32_wmma<256><<<dim3(MtPairs, 4), blk, 0, stream>>>(mid, WTout, b_out, out, L);

    (void)n_in; (void)out_size; (void)ws_size;
}